// TopkCommunityUpdater_83992380440994
// MI455X (gfx1250) — compile-verified
//
#include <hip/hip_runtime.h>

typedef __attribute__((ext_vector_type(2))) float v2f;
typedef __attribute__((ext_vector_type(8))) float v8f;

// ---- prologue: active-community flag table in d_ws ----
__global__ void zero_flags_kernel(int* __restrict__ flags, int n) {
    int i = blockIdx.x * blockDim.x + threadIdx.x;
    if (i < n) flags[i] = 0;
}

__global__ void mark_flags_kernel(const int* __restrict__ community_index,
                                  int* __restrict__ flags, int n_active) {
    int i = blockIdx.x * blockDim.x + threadIdx.x;
    if (i < n_active) flags[community_index[i]] = 1;
}

// ---- main: masked rank-1 outer product via V_WMMA_F32_16X16X4_F32 ----
// grid = B blocks, block = 32 * (M/16) threads (one wave per 16-member slab).
__global__ __launch_bounds__(256)
void outer_product_kernel(const float* __restrict__ unique_message,  // [B, D]
                          const float* __restrict__ member_score,    // [NC, M]
                          const float* __restrict__ timestamps,      // [B]
                          const int*   __restrict__ nodes,           // [B]
                          const int*   __restrict__ node2community,  // [NN]
                          const int*   __restrict__ community2node,  // [NC, M]
                          const int*   __restrict__ member_num,      // [NC]
                          const int*   __restrict__ flags,           // [NC]
                          float*       __restrict__ out_messages,    // [B, M, D]
                          int*         __restrict__ out_nodes,       // [B, M]
                          float*       __restrict__ out_ts,          // [B, M]
                          int B, int D, int M) {
    const int b      = blockIdx.x;
    const int wave   = threadIdx.x >> 5;   // member-block index (m0 = wave*16)
    const int lane   = threadIdx.x & 31;
    const int lane16 = lane & 15;
    const int m0     = wave << 4;

    // Wave-uniform event metadata (compiler scalarizes: blockIdx-based address).
    const int   node = nodes[b];
    const int   cid  = node2community[node];
    const int   use  = flags[cid];
    const int   mnum = member_num[cid];
    const float ts   = timestamps[b];

    // Lanes 0-15 carry the 16 member rows of this slab.
    const int  m_idx = m0 + lane16;
    const bool valid = (use != 0) && (m_idx < mnum);
    const float score = member_score[(size_t)cid * M + m_idx];

    // A matrix (16x4 f32): K=0 slot = masked scores, all other K slots zero.
    v2f A;
    A.x = (lane < 16 && valid) ? score : 0.0f;  // VGPR0: lanes0-15=K0, lanes16-31=K2
    A.y = 0.0f;                                 // VGPR1: K1 / K3

    // Side outputs (raw int32 bits into the f32-typed region for upd_nodes).
    if (lane < 16) {
        const int cn = community2node[(size_t)cid * M + m_idx];
        out_nodes[(size_t)b * M + m_idx] = valid ? cn : -1;
        out_ts   [(size_t)b * M + m_idx] = valid ? ts : 0.0f;
    }

    const float* __restrict__ msg   = unique_message + (size_t)b * D;
    float*       __restrict__ obase = out_messages + ((size_t)b * M + m0) * D;

    const int ntiles = D >> 4;
    const int rowadd = (lane < 16) ? 0 : 8;  // C/D layout: VGPRr -> row r / r+8
    for (int dt = 0; dt < ntiles; ++dt) {
        const int d0 = dt << 4;
        const float mv = msg[d0 + lane16];
        // B matrix (4x16 f32): replicate msg row into ALL K slots -> result is
        // score[m]*msg[n] regardless of which physical slot is K=0 (A has a
        // single nonzero K slot, so every other product term is 0).
        v2f Bm; Bm.x = mv; Bm.y = mv;
        v8f C = {};
        // 8 args: (neg_a, A, neg_b, B, c_mod, C, reuse_a, reuse_b)
        v8f Dt = __builtin_amdgcn_wmma_f32_16x16x4_f32(
            false, A, false, Bm, (short)0, C, false, false);
#pragma unroll
        for (int r = 0; r < 8; ++r) {
            // 512MB streaming output >> 192MB L2: bypass with non-temporal stores.
            __builtin_nontemporal_store(
                Dt[r], &obase[(size_t)(r + rowadd) * D + d0 + lane16]);
        }
    }
}

extern "C" void kernel_launch(void* const* d_in, const int* in_sizes, int n_in,
                              void* d_out, int out_size, void* d_ws, size_t ws_size,
                              hipStream_t stream) {
    const float* unique_message  = (const float*)d_in[0];
    const float* member_score    = (const float*)d_in[1];
    const float* timestamps      = (const float*)d_in[2];
    const int*   nodes           = (const int*)d_in[3];
    const int*   node2community  = (const int*)d_in[4];
    const int*   community2node  = (const int*)d_in[5];
    const int*   member_num      = (const int*)d_in[6];
    const int*   community_index = (const int*)d_in[7];

    const int B     = in_sizes[2];
    const int D     = in_sizes[0] / B;          // 256
    const int NCOMM = in_sizes[6];              // 2000
    const int M     = in_sizes[1] / NCOMM;      // 128
    const int NACT  = in_sizes[7];              // 1500

    int* flags = (int*)d_ws;                    // [NCOMM] scratch

    float*       out_messages = (float*)d_out;
    const size_t msg_elems    = (size_t)B * M * D;
    int*         out_nodes    = (int*)((float*)d_out + msg_elems);
    float*       out_ts       = (float*)d_out + msg_elems + (size_t)B * M;

    zero_flags_kernel<<<(NCOMM + 255) / 256, 256, 0, stream>>>(flags, NCOMM);
    mark_flags_kernel<<<(NACT + 255) / 256, 256, 0, stream>>>(community_index, flags, NACT);

    const int threads = 32 * (M / 16);          // 256 for M=128 (8 waves/block)
    outer_product_kernel<<<B, threads, 0, stream>>>(
        unique_message, member_score, timestamps, nodes, node2community,
        community2node, member_num, flags, out_messages, out_nodes, out_ts,
        B, D, M);
}